// HolomorphicSymbolicSynthesizer_2980707303740
// MI455X (gfx1250) — compile-verified
//
#include <hip/hip_runtime.h>
#include <math.h>

#define SEQ_LEN   524288
#define LATENT    256
#define CHUNK     64
#define NCHUNK    (SEQ_LEN / CHUNK)     // 8192 chunks
#define P1_BLOCK  256
#define P1_GRID   (NCHUNK / P1_BLOCK)   // 32 blocks
#define SCAN_T    256
#define SER       (NCHUNK / SCAN_T)     // 32 serial combines per scan thread
#define RED_BLOCKS 256
#define EPSQ      1e-12f

typedef __attribute__((ext_vector_type(2))) float v2f;
typedef __attribute__((ext_vector_type(8))) float v8f;
typedef int v4i __attribute__((vector_size(16)));   // matches builtin param type

// Quaternion as float4: x=w, y=x, z=y, w=z  (component 0..3 like the reference)
__device__ __forceinline__ float4 qmul(float4 a, float4 b) {
    return make_float4(
        a.x*b.x - a.y*b.y - a.z*b.z - a.w*b.w,
        a.x*b.y + a.y*b.x + a.z*b.w - a.w*b.z,
        a.x*b.z - a.y*b.w + a.z*b.x + a.w*b.y,
        a.x*b.w + a.y*b.z - a.z*b.y + a.w*b.x);
}

__device__ __forceinline__ float4 qnorm(float4 q) {
    float n = sqrtf(q.x*q.x + q.y*q.y + q.z*q.z + q.w*q.w) + EPSQ;
    float inv = 1.0f / n;
    return make_float4(q.x*inv, q.y*inv, q.z*inv, q.w*inv);
}

__device__ __forceinline__ float4 qident() { return make_float4(1.f, 0.f, 0.f, 0.f); }

// Broadcast the 256x4 rot table into LDS. Prefer the gfx1250 async
// global->LDS path (ASYNCcnt) when the toolchain exposes it.
__device__ __forceinline__ void load_table_to_lds(const float4* __restrict__ table,
                                                  float4* lt) {
#if __has_builtin(__builtin_amdgcn_global_load_async_to_lds_b128) && \
    __has_builtin(__builtin_amdgcn_s_wait_asynccnt)
    if (threadIdx.x < LATENT) {
        __builtin_amdgcn_global_load_async_to_lds_b128(
            (__attribute__((address_space(1))) v4i*)(table + threadIdx.x),
            (__attribute__((address_space(3))) v4i*)(lt + threadIdx.x),
            /*offset=*/0, /*cpol=*/0);
    }
    __builtin_amdgcn_s_wait_asynccnt(0);
#else
    if (threadIdx.x < LATENT) lt[threadIdx.x] = table[threadIdx.x];
#endif
    __syncthreads();
}

// ---------------------------------------------------------------------------
// Kernel A: rot_table[256][4] = E(256x4) @ W^T(4x4) via V_WMMA_F32_16X16X4_F32
// One block of 512 threads = 16 waves; wave w computes rows [16w, 16w+16).
// Branchless operand construction so EXEC is all-1s at the WMMA.
// ---------------------------------------------------------------------------
__global__ void rot_table_wmma(const float* __restrict__ E,
                               const float* __restrict__ W,
                               float* __restrict__ table) {
    int tid   = threadIdx.x;
    int wave  = tid >> 5;
    int lane  = tid & 31;
    int n     = lane & 15;
    int khalf = lane >> 4;          // 0: K=0,1   1: K=2,3
    int arow  = wave * 16 + n;      // A-matrix M index for this lane

    v2f a, b;
    a[0] = E[arow * 4 + 2 * khalf + 0];
    a[1] = E[arow * 4 + 2 * khalf + 1];
    // B = W^T zero-padded to 4x16: B[k][n] = W[n][k] for n<4 else 0.
    // Unconditional in-bounds load + VALU select (no EXEC divergence).
    float w0 = W[(n & 3) * 4 + 2 * khalf + 0];
    float w1 = W[(n & 3) * 4 + 2 * khalf + 1];
    b[0] = (n < 4) ? w0 : 0.0f;
    b[1] = (n < 4) ? w1 : 0.0f;

    v8f c = {};
    c = __builtin_amdgcn_wmma_f32_16x16x4_f32(
            /*neg_a=*/false, a, /*neg_b=*/false, b,
            /*c_mod=*/(short)0, c, /*reuse_a=*/false, /*reuse_b=*/false);

    if (n < 4) {
        int mbase = khalf * 8;
#pragma unroll
        for (int i = 0; i < 8; ++i) {
            int row = wave * 16 + mbase + i;
            table[row * 4 + n] = c[i];
        }
    }
}

// ---------------------------------------------------------------------------
// Kernel B (phase 1): per-chunk quaternion product with per-step normalize.
// ---------------------------------------------------------------------------
__global__ void phase1_chunks(const int* __restrict__ sym,
                              const float4* __restrict__ table,
                              float4* __restrict__ chunk_prod) {
    __shared__ float4 lt[LATENT];
    int t    = blockIdx.x * blockDim.x + threadIdx.x;
    int base = t * CHUNK;
    __builtin_prefetch(sym + base, 0, 1);
    __builtin_prefetch(sym + base + 32, 0, 1);
    load_table_to_lds(table, lt);

    float4 q = qident();
#pragma unroll 4
    for (int i = 0; i < CHUNK; ++i) {
        int s = sym[base + i];
        q = qnorm(qmul(q, lt[s]));
    }
    chunk_prod[t] = q;
}

// ---------------------------------------------------------------------------
// Kernel C (phase 2): exclusive scan of 8192 chunk products (one block).
// 256 threads x 32 serial, then Hillis-Steele over 256 aggregates in LDS.
// ---------------------------------------------------------------------------
__global__ void scan_chunks(const float4* __restrict__ cp,
                            float4* __restrict__ cpre) {
    __shared__ float4 bufA[SCAN_T];
    __shared__ float4 bufB[SCAN_T];
    int j    = threadIdx.x;
    int base = j * SER;

    float4 p = qident();
#pragma unroll 4
    for (int i = 0; i < SER; ++i) p = qnorm(qmul(p, cp[base + i]));
    bufA[j] = p;
    __syncthreads();

    float4* src = bufA;
    float4* dst = bufB;
    for (int ofs = 1; ofs < SCAN_T; ofs <<= 1) {
        float4 v = src[j];
        if (j >= ofs) v = qnorm(qmul(src[j - ofs], v));
        dst[j] = v;
        __syncthreads();
        float4* tmp = src; src = dst; dst = tmp;
    }

    float4 q = (j == 0) ? qident() : src[j - 1];   // exclusive prefix
#pragma unroll 4
    for (int i = 0; i < SER; ++i) {
        cpre[base + i] = q;
        q = qnorm(qmul(q, cp[base + i]));
    }
}

// ---------------------------------------------------------------------------
// Kernel D (phase 3): replay each chunk from its exclusive prefix, emit q_t.
// ---------------------------------------------------------------------------
__global__ void phase3_emit(const int* __restrict__ sym,
                            const float4* __restrict__ table,
                            const float4* __restrict__ cpre,
                            float4* __restrict__ out) {
    __shared__ float4 lt[LATENT];
    int t    = blockIdx.x * blockDim.x + threadIdx.x;
    int base = t * CHUNK;
    __builtin_prefetch(sym + base, 0, 1);
    __builtin_prefetch(sym + base + 32, 0, 1);
    load_table_to_lds(table, lt);

    float4 q = cpre[t];
#pragma unroll 4
    for (int i = 0; i < CHUNK; ++i) {
        int s = sym[base + i];
        q = qnorm(qmul(q, lt[s]));
        out[base + i] = q;
    }
}

// ---------------------------------------------------------------------------
// Kernel E: biharmonic stencil partial sums (fixed grid -> deterministic).
// ---------------------------------------------------------------------------
__global__ void biharm_partial(const float4* __restrict__ q,
                               float* __restrict__ partial) {
    __shared__ float red[256];
    float acc = 0.0f;
    int stride = gridDim.x * blockDim.x;
    for (int t = blockIdx.x * blockDim.x + threadIdx.x; t < SEQ_LEN - 4; t += stride) {
        float4 a = q[t], b = q[t + 1], c = q[t + 2], d = q[t + 3], e = q[t + 4];
        float bx = a.x - 4.f*b.x + 6.f*c.x - 4.f*d.x + e.x;
        float by = a.y - 4.f*b.y + 6.f*c.y - 4.f*d.y + e.y;
        float bz = a.z - 4.f*b.z + 6.f*c.z - 4.f*d.z + e.z;
        float bw = a.w - 4.f*b.w + 6.f*c.w - 4.f*d.w + e.w;
        acc += bx*bx + by*by + bz*bz + bw*bw;
    }
    red[threadIdx.x] = acc;
    __syncthreads();
    for (int s = 128; s > 0; s >>= 1) {
        if (threadIdx.x < s) red[threadIdx.x] += red[threadIdx.x + s];
        __syncthreads();
    }
    if (threadIdx.x == 0) partial[blockIdx.x] = red[0];
}

// ---------------------------------------------------------------------------
// Kernel F: final reduce -> curvature_tear scalar + scale factor.
// ---------------------------------------------------------------------------
__global__ void finalize_ct(const float* __restrict__ partial,
                            float* __restrict__ out_ct,
                            float* __restrict__ scale) {
    __shared__ float red[RED_BLOCKS];
    red[threadIdx.x] = partial[threadIdx.x];
    __syncthreads();
    for (int s = RED_BLOCKS / 2; s > 0; s >>= 1) {
        if (threadIdx.x < s) red[threadIdx.x] += red[threadIdx.x + s];
        __syncthreads();
    }
    if (threadIdx.x == 0) {
        float ct = sqrtf(red[0]);
        out_ct[0] = ct;
        scale[0]  = (ct > 1e-3f) ? (1.0f / (1.0f + ct)) : 1.0f;
    }
}

// ---------------------------------------------------------------------------
// Kernel G: conditional rescale of the whole surface.
// ---------------------------------------------------------------------------
__global__ void apply_scale(float4* __restrict__ q,
                            const float* __restrict__ scale) {
    float s = scale[0];
    int i = blockIdx.x * blockDim.x + threadIdx.x;
    if (i < SEQ_LEN) {
        float4 v = q[i];
        q[i] = make_float4(v.x * s, v.y * s, v.z * s, v.w * s);
    }
}

extern "C" void kernel_launch(void* const* d_in, const int* in_sizes, int n_in,
                              void* d_out, int out_size, void* d_ws, size_t ws_size,
                              hipStream_t stream) {
    (void)in_sizes; (void)n_in; (void)out_size; (void)ws_size;
    const int*   sym = (const int*)d_in[0];
    const float* E   = (const float*)d_in[1];
    const float* W   = (const float*)d_in[2];
    float*       out = (float*)d_out;

    char* ws = (char*)d_ws;
    float*  table       = (float*)(ws);                                   //   4 KB
    float4* chunk_prod  = (float4*)(ws + 4096);                           // 128 KB
    float4* chunk_pref  = (float4*)(ws + 4096 + (size_t)NCHUNK * 16);     // 128 KB
    float*  partial     = (float*)(ws + 4096 + 2 * (size_t)NCHUNK * 16);  //   1 KB
    float*  scale       = partial + RED_BLOCKS;

    rot_table_wmma<<<1, 512, 0, stream>>>(E, W, table);
    phase1_chunks<<<P1_GRID, P1_BLOCK, 0, stream>>>(sym, (const float4*)table, chunk_prod);
    scan_chunks<<<1, SCAN_T, 0, stream>>>(chunk_prod, chunk_pref);
    phase3_emit<<<P1_GRID, P1_BLOCK, 0, stream>>>(sym, (const float4*)table, chunk_pref, (float4*)out);
    biharm_partial<<<RED_BLOCKS, 256, 0, stream>>>((const float4*)out, partial);
    finalize_ct<<<1, RED_BLOCKS, 0, stream>>>(partial, out + (size_t)SEQ_LEN * 4, scale);
    apply_scale<<<SEQ_LEN / 256, 256, 0, stream>>>((float4*)out, scale);
}